// IntraRegionLoss_17179869184487
// MI455X (gfx1250) — compile-verified
//
#include <hip/hip_runtime.h>
#include <hip/hip_bf16.h>

// IntraRegionLoss: two masked softmax-CE reductions over [64,1024,1024] fp32
// logits. Memory-bound: 512 MB read-once => ~22us at 23.3 TB/s. One wave32 per
// row, coalesced b128 loads, fp32 throughout, wave exp-sum reduced via a chain
// of V_WMMA_F32_16X16X4_F32, block partials + tiny finalize (no atomics).

typedef float v2f __attribute__((ext_vector_type(2)));
typedef float v8f __attribute__((ext_vector_type(8)));

#ifndef __has_builtin
#define __has_builtin(x) 0
#endif

constexpr int kB           = 64;
constexpr int kN           = 1024;
constexpr int kRowsPerDir  = kB * kN;          // 65536 rows per direction
constexpr int kTotalRows   = 2 * kRowsPerDir;  // 131072
constexpr int kWavesPerBlk = 8;                // 256 threads = 8 wave32
constexpr int kBlocks      = kTotalRows / kWavesPerBlk;  // 16384

// Sum a per-lane fp32 value across all 32 lanes using WMMA f32 16x16x4.
// Requires EXEC == all ones (guaranteed by caller: full waves, uniform flow).
//
// Step 1: A[M][K] = {s_M, s_M, s_{M+16}, s_{M+16}} (documented A layout:
//         lanes 0-15 hold K=0/1, lanes 16-31 hold K=2/3), B = 0.5 everywhere:
//         D1[M][*] = s_M + s_{M+16} = t_M.
// Step 2: four accumulator-chained WMMAs with A = ones and B = consecutive
//         D1 VGPR pairs. With A = ones, D += sum of the 4 B values in each
//         column, independent of their K-slot assignment, so the chain sums
//         all 16 t_M regardless of the exact B K-layout. Result S in every
//         lane of acc[0].
__device__ __forceinline__ float wave_sum32(float s) {
#if __has_builtin(__builtin_amdgcn_wmma_f32_16x16x4_f32)
  v2f a;    a[0] = s;    a[1] = s;
  v2f half; half[0] = 0.5f; half[1] = 0.5f;
  v8f z = {0.f, 0.f, 0.f, 0.f, 0.f, 0.f, 0.f, 0.f};
  v8f d1 = __builtin_amdgcn_wmma_f32_16x16x4_f32(
      false, a, false, half, (short)0, z, false, false);
  v2f ones; ones[0] = 1.0f; ones[1] = 1.0f;
  v8f acc = z;
  v2f b;
  b[0] = d1[0]; b[1] = d1[1];
  acc = __builtin_amdgcn_wmma_f32_16x16x4_f32(false, ones, false, b, (short)0, acc, false, false);
  b[0] = d1[2]; b[1] = d1[3];
  acc = __builtin_amdgcn_wmma_f32_16x16x4_f32(false, ones, false, b, (short)0, acc, false, false);
  b[0] = d1[4]; b[1] = d1[5];
  acc = __builtin_amdgcn_wmma_f32_16x16x4_f32(false, ones, false, b, (short)0, acc, false, false);
  b[0] = d1[6]; b[1] = d1[7];
  acc = __builtin_amdgcn_wmma_f32_16x16x4_f32(false, ones, false, b, (short)0, acc, false, false);
  return acc[0];
#else
  #pragma unroll
  for (int off = 16; off > 0; off >>= 1) s += __shfl_xor(s, off, 32);
  return s;
#endif
}

__global__ __launch_bounds__(256) void IntraRegionLoss_row_kernel(
    const float* __restrict__ succ_logits,
    const int*   __restrict__ succ_labels,
    const float* __restrict__ pred_logits,
    const int*   __restrict__ pred_labels,
    const int*   __restrict__ line_mask,
    float* __restrict__ block_partials /* [kBlocks][2] */) {
  __shared__ float s_nll[kWavesPerBlk];
  __shared__ int   s_dir[kWavesPerBlk];

  const int tid  = threadIdx.x;
  const int lane = tid & 31;
  const int wave = tid >> 5;
  const int row  = blockIdx.x * kWavesPerBlk + wave;   // [0, kTotalRows)
  const int dir  = (row >= kRowsPerDir) ? 1 : 0;       // 0 = succ, 1 = pred
  const int idx  = dir ? (row - kRowsPerDir) : row;    // b*N + n
  const int i    = idx & (kN - 1);                     // line index within batch

  const float* __restrict__ logits =
      (dir ? pred_logits : succ_logits) + (size_t)idx * kN;
  const int* __restrict__ labels = dir ? pred_labels : succ_labels;

  // Stream the 4 KB row once: 8 x float4 per lane, coalesced 512B/issue.
  const float4* rp = (const float4*)logits;
  float4 v[8];
#pragma unroll
  for (int c = 0; c < 8; ++c) v[c] = rp[c * 32 + lane];

  // Pass 1 (registers): row max.
  float m = -3.402823466e+38f;
#pragma unroll
  for (int c = 0; c < 8; ++c)
    m = fmaxf(m, fmaxf(fmaxf(v[c].x, v[c].y), fmaxf(v[c].z, v[c].w)));
#pragma unroll
  for (int off = 16; off > 0; off >>= 1) m = fmaxf(m, __shfl_xor(m, off, 32));

  // Per-row metadata (wave-uniform).
  const int  lraw  = labels[idx];
  const int  maskv = (line_mask[idx] != 0) ? 1 : 0;
  const bool selfp = (lraw == -1);
  const bool cond  = selfp && (maskv != 0);        // diagonal gets row_max+1
  const float madj = cond ? (m + 1.0f) : m;        // max of adjusted row
  const int  lbl   = selfp ? i : min(max(lraw, 0), kN - 1);

  // Pass 2 (registers): sum exp(x - madj); args are always <= 0.
  float s = 0.0f;
#pragma unroll
  for (int c = 0; c < 8; ++c) {
    s += __expf(v[c].x - madj);
    s += __expf(v[c].y - madj);
    s += __expf(v[c].z - madj);
    s += __expf(v[c].w - madj);
  }

  // Wave reduction via WMMA chain (EXEC all ones here: full waves, no
  // divergent branches above, per-wave-uniform selects only).
  float S = wave_sum32(s);

  // Uniform diagonal correction: replace exp(x_ii - madj) by exp(0) = 1.
  const float x_diag = logits[i];
  S = cond ? (S - __expf(x_diag - madj) + 1.0f) : S;

  const float x_lbl  = logits[lbl];
  const float chosen = cond ? madj : x_lbl;        // adjusted logit at label
  float nll = madj + __logf(S) - chosen;           // logsumexp(adj) - chosen
  nll *= (float)maskv;

  if (lane == 0) { s_nll[wave] = nll; s_dir[wave] = dir; }
  __syncthreads();
  if (tid < 2) {  // tid 0 sums succ waves, tid 1 sums pred waves
    float acc = 0.0f;
#pragma unroll
    for (int w = 0; w < kWavesPerBlk; ++w)
      if (s_dir[w] == tid) acc += s_nll[w];
    block_partials[blockIdx.x * 2 + tid] = acc;
  }
}

__global__ __launch_bounds__(256) void IntraRegionLoss_finalize_kernel(
    const int*   __restrict__ line_mask,
    const float* __restrict__ pred_weight,
    const float* __restrict__ block_partials,
    float* __restrict__ out) {
  __shared__ float sh_s[256];
  __shared__ float sh_p[256];
  __shared__ float sh_m[256];
  const int t = threadIdx.x;

  float ss = 0.0f, pp = 0.0f, mv = 0.0f;
  for (int b = t; b < kBlocks; b += 256) {
    ss += block_partials[2 * b + 0];
    pp += block_partials[2 * b + 1];
  }
  for (int k = t; k < kRowsPerDir; k += 256)
    mv += (line_mask[k] != 0) ? 1.0f : 0.0f;

  sh_s[t] = ss; sh_p[t] = pp; sh_m[t] = mv;
  __syncthreads();
#pragma unroll
  for (int stride = 128; stride > 0; stride >>= 1) {
    if (t < stride) {
      sh_s[t] += sh_s[t + stride];
      sh_p[t] += sh_p[t + stride];
      sh_m[t] += sh_m[t + stride];
    }
    __syncthreads();
  }
  if (t == 0) {
    const float valid = sh_m[0];
    const float denom = fmaxf(valid, 1.0f);
    const float succ  = sh_s[0] / denom;
    const float pred  = sh_p[0] / denom;
    const float w     = pred_weight[0];
    out[0] = succ + w * pred;  // total_loss
    out[1] = succ;             // succ_loss
    out[2] = pred;             // pred_loss
    out[3] = valid;            // num_valid
  }
}

extern "C" void kernel_launch(void* const* d_in, const int* in_sizes, int n_in,
                              void* d_out, int out_size, void* d_ws,
                              size_t ws_size, hipStream_t stream) {
  const float* succ_logits = (const float*)d_in[0];
  const int*   succ_labels = (const int*)d_in[1];
  const float* pred_logits = (const float*)d_in[2];
  const int*   pred_labels = (const int*)d_in[3];
  const int*   line_mask   = (const int*)d_in[4];
  const float* pred_weight = (const float*)d_in[5];
  float* partials = (float*)d_ws;  // kBlocks*2 floats = 128 KB, fully
                                   // rewritten every call (deterministic)

  IntraRegionLoss_row_kernel<<<kBlocks, 256, 0, stream>>>(
      succ_logits, succ_labels, pred_logits, pred_labels, line_mask, partials);
  IntraRegionLoss_finalize_kernel<<<1, 256, 0, stream>>>(
      line_mask, pred_weight, partials, (float*)d_out);
}